// DifferentiableKabsch_56831007261024
// MI455X (gfx1250) — compile-verified
//
#include <hip/hip_runtime.h>

typedef __attribute__((ext_vector_type(2))) float v2f;
typedef __attribute__((ext_vector_type(4))) float v4f;
typedef __attribute__((ext_vector_type(8))) float v8f;

#define KEPS 1e-6f

__device__ __forceinline__ v4f ldnt4(const float* p) {
  return __builtin_nontemporal_load((const v4f*)p);
}

__global__ __launch_bounds__(256, 2) void kabsch_kernel(
    const float* __restrict__ src, const float* __restrict__ tgt,
    const float* __restrict__ wgt, float* __restrict__ outR,
    float* __restrict__ outT, int N)
{
  constexpr int TPB = 256;
  constexpr int RS  = 17;                 // padded row stride -> no LDS bank conflicts
  __shared__ float smPart[TPB * RS];      // per-thread 16 partial moments
  __shared__ float smSums[16];            // reduced moments

  const int b   = blockIdx.x;
  const int tid = threadIdx.x;

  const float* sB = src + (size_t)b * N * 3;
  const float* tB = tgt + (size_t)b * N * 3;
  const float* wB = wgt + (size_t)b * N;

  // ---- streaming moment accumulation (memory-bound part) ----
  float accW = 0.f;
  float accS[3] = {0.f, 0.f, 0.f};
  float accT[3] = {0.f, 0.f, 0.f};
  float accM[9] = {0.f,0.f,0.f,0.f,0.f,0.f,0.f,0.f,0.f};

  for (int i = 0; i < N; i += TPB * 4) {
    const int p0 = i + tid * 4;           // 4 points per thread per iter
    v4f w4 = ldnt4(wB + p0);
    const float* sp = sB + 3 * p0;        // 48-byte aligned
    const float* tp = tB + 3 * p0;
    v4f sa = ldnt4(sp), sb2 = ldnt4(sp + 4), sc = ldnt4(sp + 8);
    v4f ta = ldnt4(tp), tb2 = ldnt4(tp + 4), tc = ldnt4(tp + 8);

    float ss[12] = {sa.x, sa.y, sa.z, sa.w, sb2.x, sb2.y, sb2.z, sb2.w,
                    sc.x, sc.y, sc.z, sc.w};
    float tt[12] = {ta.x, ta.y, ta.z, ta.w, tb2.x, tb2.y, tb2.z, tb2.w,
                    tc.x, tc.y, tc.z, tc.w};
    float ww[4]  = {w4.x, w4.y, w4.z, w4.w};

#pragma unroll
    for (int k = 0; k < 4; ++k) {
      const float w  = ww[k];
      const float sx = ss[3*k], sy = ss[3*k+1], sz = ss[3*k+2];
      const float tx = tt[3*k], ty = tt[3*k+1], tz = tt[3*k+2];
      const float wsx = w * sx, wsy = w * sy, wsz = w * sz;
      accW += w;
      accS[0] += wsx;  accS[1] += wsy;  accS[2] += wsz;
      accT[0] = fmaf(w, tx, accT[0]);
      accT[1] = fmaf(w, ty, accT[1]);
      accT[2] = fmaf(w, tz, accT[2]);
      accM[0] = fmaf(wsx, tx, accM[0]);
      accM[1] = fmaf(wsx, ty, accM[1]);
      accM[2] = fmaf(wsx, tz, accM[2]);
      accM[3] = fmaf(wsy, tx, accM[3]);
      accM[4] = fmaf(wsy, ty, accM[4]);
      accM[5] = fmaf(wsy, tz, accM[5]);
      accM[6] = fmaf(wsz, tx, accM[6]);
      accM[7] = fmaf(wsz, ty, accM[7]);
      accM[8] = fmaf(wsz, tz, accM[8]);
    }
  }

  // stage 16 partial moments per thread into LDS (row-padded)
  {
    float* row = &smPart[tid * RS];
#pragma unroll
    for (int j = 0; j < 9; ++j) row[j] = accM[j];
    row[9]  = accS[0]; row[10] = accS[1]; row[11] = accS[2];
    row[12] = accT[0]; row[13] = accT[1]; row[14] = accT[2];
    row[15] = accW;
  }
  __syncthreads();

  // ---- WMMA f32 reduction: C = ones(16x4) x partials(4x16) accumulated ----
  // A-matrix 16x4 all-ones; B-matrix 4x16 holds 4 partial rows x 16 moments.
  // Row 0 of C ends with sums[j] in lane j. Two accumulators hide WMMA RAW.
  if (tid < 32) {
    v8f c0 = {0.f,0.f,0.f,0.f,0.f,0.f,0.f,0.f};
    v8f c1 = {0.f,0.f,0.f,0.f,0.f,0.f,0.f,0.f};
    v2f a;  a.x = 1.f; a.y = 1.f;
    const int col = tid & 15;             // moment column
    const int k0  = (tid >> 4) << 1;      // lanes 0-15 hold K=0,1; 16-31 hold K=2,3
#pragma unroll 4
    for (int g = 0; g < TPB / 4; g += 2) {
      v2f b0, b1;
      b0.x = smPart[(g * 4 + k0    ) * RS + col];
      b0.y = smPart[(g * 4 + k0 + 1) * RS + col];
      b1.x = smPart[((g + 1) * 4 + k0    ) * RS + col];
      b1.y = smPart[((g + 1) * 4 + k0 + 1) * RS + col];
      c0 = __builtin_amdgcn_wmma_f32_16x16x4_f32(false, a, false, b0,
                                                 (short)0, c0, false, false);
      c1 = __builtin_amdgcn_wmma_f32_16x16x4_f32(false, a, false, b1,
                                                 (short)0, c1, false, false);
    }
    if (tid < 16) smSums[tid] = c0[0] + c1[0];   // C row M=0, N=lane
  }
  __syncthreads();

  // ---- per-batch 3x3 Kabsch solve on thread 0 ----
  if (tid == 0) {
    const float W   = smSums[15];
    const float inv = 1.0f / (W + KEPS);
    float cs[3] = {smSums[9]  * inv, smSums[10] * inv, smSums[11] * inv};
    float ct[3] = {smSums[12] * inv, smSums[13] * inv, smSums[14] * inv};
    const float kf = 2.0f - W * inv;

    // H = M/(W+eps) - kf * cs ct^T + eps*I
    float H[3][3];
#pragma unroll
    for (int d = 0; d < 3; ++d)
#pragma unroll
      for (int e = 0; e < 3; ++e)
        H[d][e] = smSums[d * 3 + e] * inv - kf * cs[d] * ct[e];
    H[0][0] += KEPS; H[1][1] += KEPS; H[2][2] += KEPS;

    // A = H^T H (symmetric PSD), Jacobi eigen -> V, lambda
    float A[3][3];
#pragma unroll
    for (int i = 0; i < 3; ++i)
#pragma unroll
      for (int j = 0; j < 3; ++j)
        A[i][j] = H[0][i] * H[0][j] + H[1][i] * H[1][j] + H[2][i] * H[2][j];

    float Vv[3][3] = {{1.f,0.f,0.f},{0.f,1.f,0.f},{0.f,0.f,1.f}};
    const int PP[3] = {0, 0, 1};
    const int QQ[3] = {1, 2, 2};
    for (int sweep = 0; sweep < 8; ++sweep) {
#pragma unroll
      for (int m = 0; m < 3; ++m) {
        const int p = PP[m], q = QQ[m], r = 3 - p - q;
        const float apq = A[p][q];
        if (fabsf(apq) > 1e-20f) {
          const float theta = 0.5f * (A[q][q] - A[p][p]) / apq;
          const float tJ = copysignf(1.0f / (fabsf(theta) +
                              sqrtf(theta * theta + 1.0f)), theta);
          const float cJ = 1.0f / sqrtf(tJ * tJ + 1.0f);
          const float sJ = tJ * cJ;
          const float app = A[p][p], aqq = A[q][q];
          const float arp = A[r][p], arq = A[r][q];
          A[p][p] = app - tJ * apq;
          A[q][q] = aqq + tJ * apq;
          A[p][q] = 0.f; A[q][p] = 0.f;
          A[r][p] = cJ * arp - sJ * arq; A[p][r] = A[r][p];
          A[r][q] = sJ * arp + cJ * arq; A[q][r] = A[r][q];
#pragma unroll
          for (int rr = 0; rr < 3; ++rr) {
            const float vp = Vv[rr][p], vq = Vv[rr][q];
            Vv[rr][p] = cJ * vp - sJ * vq;
            Vv[rr][q] = sJ * vp + cJ * vq;
          }
        }
      }
    }

    // sort eigenvalues descending
    float lam[3] = {A[0][0], A[1][1], A[2][2]};
    int i0 = 0, i1 = 1, i2 = 2, tmp;
    if (lam[i0] < lam[i1]) { tmp = i0; i0 = i1; i1 = tmp; }
    if (lam[i0] < lam[i2]) { tmp = i0; i0 = i2; i2 = tmp; }
    if (lam[i1] < lam[i2]) { tmp = i1; i1 = i2; i2 = tmp; }

    float v0[3] = {Vv[0][i0], Vv[1][i0], Vv[2][i0]};
    float v1[3] = {Vv[0][i1], Vv[1][i1], Vv[2][i1]};

    // u_i = H v_i / sigma_i  (normalize; re-orthogonalize u1)
    float u0[3], u1[3];
#pragma unroll
    for (int d = 0; d < 3; ++d) {
      u0[d] = H[d][0] * v0[0] + H[d][1] * v0[1] + H[d][2] * v0[2];
      u1[d] = H[d][0] * v1[0] + H[d][1] * v1[1] + H[d][2] * v1[2];
    }
    float n0 = 1.0f / sqrtf(u0[0]*u0[0] + u0[1]*u0[1] + u0[2]*u0[2] + 1e-30f);
    u0[0] *= n0; u0[1] *= n0; u0[2] *= n0;
    const float d01 = u0[0]*u1[0] + u0[1]*u1[1] + u0[2]*u1[2];
    u1[0] -= d01 * u0[0]; u1[1] -= d01 * u0[1]; u1[2] -= d01 * u0[2];
    float n1 = 1.0f / sqrtf(u1[0]*u1[0] + u1[1]*u1[1] + u1[2]*u1[2] + 1e-30f);
    u1[0] *= n1; u1[1] *= n1; u1[2] *= n1;

    // third axes via cross products => reflection correction folded in:
    // R = [v0 v1 v0xv1] [u0 u1 u0xu1]^T  ==  V diag(1,1,sign det) U^T
    float u2[3] = {u0[1]*u1[2] - u0[2]*u1[1],
                   u0[2]*u1[0] - u0[0]*u1[2],
                   u0[0]*u1[1] - u0[1]*u1[0]};
    float v2[3] = {v0[1]*v1[2] - v0[2]*v1[1],
                   v0[2]*v1[0] - v0[0]*v1[2],
                   v0[0]*v1[1] - v0[1]*v1[0]};

    float R[3][3];
#pragma unroll
    for (int i = 0; i < 3; ++i)
#pragma unroll
      for (int j = 0; j < 3; ++j)
        R[i][j] = v0[i]*u0[j] + v1[i]*u1[j] + v2[i]*u2[j];

#pragma unroll
    for (int i = 0; i < 3; ++i) {
      const float ti = ct[i] - (R[i][0]*cs[0] + R[i][1]*cs[1] + R[i][2]*cs[2]);
#pragma unroll
      for (int j = 0; j < 3; ++j) outR[(size_t)b * 9 + i * 3 + j] = R[i][j];
      outT[(size_t)b * 3 + i] = ti;
    }
  }
}

extern "C" void kernel_launch(void* const* d_in, const int* in_sizes, int n_in,
                              void* d_out, int out_size, void* d_ws, size_t ws_size,
                              hipStream_t stream) {
  (void)n_in; (void)out_size; (void)d_ws; (void)ws_size;
  const float* src = (const float*)d_in[0];
  const float* tgt = (const float*)d_in[1];
  const float* wgt = (const float*)d_in[2];
  const int N = 4096;
  const int B = in_sizes[2] / N;           // weights is (B, N)
  float* outR = (float*)d_out;             // (B,3,3) flat
  float* outT = outR + (size_t)B * 9;      // (B,3) flat
  kabsch_kernel<<<B, 256, 0, stream>>>(src, tgt, wgt, outR, outT, N);
}